// ClusteredAttentionAggregator_50852412785042
// MI455X (gfx1250) — compile-verified
//
#include <hip/hip_runtime.h>
#include <hip/hip_bf16.h>

typedef float v2f __attribute__((ext_vector_type(2)));
typedef float v4f __attribute__((ext_vector_type(4)));
typedef float v8f __attribute__((ext_vector_type(8)));

#define B_TOT   20000
#define S_N     10
#define F_DIM   128
#define K_CL    16
#define E_DIM   128
#define TB      16      // B-rows per wave (WMMA M)
#define WAVES   4       // waves per workgroup
#define ROW_PAD 132     // 128 + 4 pad -> conflict-free A-frag reads
#define Q_PAD   18      // 16 + 2 pad  -> conflict-free + 8B aligned

// ---- LDS layout (float indices; dynamic LDS starts at LDS byte offset 0) ----
#define SM_WT        (256*128)                    // weight^T [k][e]
#define SM_CENTERT   (128*16)                     // center^T [f][k]
#define SM_CMASK     (16*128)                     // cluster_mask [k][f]
#define SM_CONST     (SM_WT + SM_CENTERT + SM_CMASK + 128 + 128 + 16)
#define OFF_SELF     0
#define OFF_NEIGH0   (TB*ROW_PAD)
#define OFF_NEIGH1   (2*TB*ROW_PAD)
#define OFF_AGG      (3*TB*ROW_PAD)
#define OFF_QST      (4*TB*ROW_PAD)
#define OFF_SCR      (OFF_QST + TB*Q_PAD)
#define SM_PERWAVE   (OFF_SCR + 64)
#define SM_TOTAL     (SM_CONST + WAVES*SM_PERWAVE)

__device__ __forceinline__ void lds_fence() {
  // cross-lane LDS handoff inside one wave
  asm volatile("s_wait_dscnt 0" ::: "memory");
}
__device__ __forceinline__ void wait_async_16() {
  asm volatile("s_wait_asynccnt 0x10" ::: "memory");
}
__device__ __forceinline__ void wait_async_0() {
  asm volatile("s_wait_asynccnt 0x0" ::: "memory");
}

// One instruction stages a full 128-float row: lane l moves 16B from
// src+16B*l to LDS rowBase+16B*l, tracked by ASYNCcnt (no VGPR landing zone).
__device__ __forceinline__ void async_row_load(unsigned ldsFloatIdx,
                                               const float* src, int lane) {
  unsigned ldsByte = (ldsFloatIdx + (unsigned)lane * 4u) * 4u;
  const float* g = src + lane * 4;
  asm volatile("global_load_async_to_lds_b128 %0, %1, off"
               :: "v"(ldsByte), "v"(g) : "memory");
}

__global__ __launch_bounds__(WAVES*32, 1)
void clustered_attn_gnn_kernel(const int*   __restrict__ nodes,
                               const int*   __restrict__ neigh_idx,
                               const float* __restrict__ self_table,
                               const float* __restrict__ neigh_table,
                               const float* __restrict__ center,
                               const float* __restrict__ cluster_mask,
                               const float* __restrict__ weight,
                               const float* __restrict__ alpha,
                               float*       __restrict__ out)
{
  extern __shared__ float smem[];
  float* s_wT      = smem;                      // [256][128]  weight^T
  float* s_centerT = s_wT + SM_WT;              // [128][16]
  float* s_cmask   = s_centerT + SM_CENTERT;    // [16][128]
  float* s_aself   = s_cmask + SM_CMASK;        // [128]
  float* s_aneigh  = s_aself + 128;             // [128]
  float* s_c2      = s_aneigh + 128;            // [16]

  const int tid  = threadIdx.x;
  const int wave = tid >> 5;
  const int lane = tid & 31;

  const unsigned wvOff = (unsigned)SM_CONST + (unsigned)wave * SM_PERWAVE;
  float* selfF  = smem + wvOff + OFF_SELF;      // [16][132]
  float* aggS   = smem + wvOff + OFF_AGG;       // [16][132]
  float* qst    = smem + wvOff + OFF_QST;       // [16][18]
  float* s_n2   = smem + wvOff + OFF_SCR;       // [16]
  float* s_att  = s_n2   + 16;                  // [16]
  float* s_dslf = s_att  + 16;                  // [16]
  float* s_asum = s_dslf + 16;                  // [16]
  const unsigned nbufOff[2] = { wvOff + OFF_NEIGH0, wvOff + OFF_NEIGH1 };

  const int bBase = blockIdx.x * (WAVES*TB) + wave*TB;

  // ---- issue async gathers FIRST (overlap with constant staging below) ----
  for (int r = 0; r < TB; ++r) {
    int b = bBase + r; if (b >= B_TOT) b = B_TOT - 1;
    async_row_load(wvOff + OFF_SELF + r*ROW_PAD,
                   self_table + (size_t)nodes[b] * F_DIM, lane);
  }
  for (int r = 0; r < TB; ++r) {
    int b = bBase + r; if (b >= B_TOT) b = B_TOT - 1;
    async_row_load(nbufOff[0] + r*ROW_PAD,
                   neigh_table + (size_t)neigh_idx[b*S_N + 0] * F_DIM, lane);
  }

  // ---------------- cooperative constant staging ----------------
  s_aself[tid]  = alpha[tid];
  s_aneigh[tid] = alpha[128 + tid];
  for (int i = tid; i < K_CL*F_DIM; i += WAVES*32) {
    int k = i >> 7, f = i & 127;
    s_centerT[f*16 + k] = center[i];
    s_cmask[i]          = cluster_mask[i];
  }
  if (tid < K_CL) {
    float s = 0.f;
    for (int f = 0; f < F_DIM; ++f) { float v = center[tid*F_DIM + f]; s += v*v; }
    s_c2[tid] = s;
  }
  // weight (E=128, 2F=256) transposed into [k][e]; thread t owns row e=t
  for (int k = 0; k < 256; ++k)
    s_wT[k*128 + tid] = weight[tid*256 + k];
  __syncthreads();

  const int rr   = lane & 15;        // A-row / B-col / C-col index
  const int half = lane >> 4;
  const int koff = half * 2;         // K sub-offset for f32 16x16x4 frags
  const int hi8  = half * 8;         // C-layout row offset

  // self rows resident (16 slot-0 ops may still be in flight, in-order)
  wait_async_16();

  // self @ a_self, split K across lane halves, combine with shfl_xor(16)
  {
    const float* row = selfF + rr*ROW_PAD + half*64;
    const float* av  = s_aself + half*64;
    float d = 0.f;
    for (int f = 0; f < 64; ++f) d += row[f]*av[f];
    d += __shfl_xor(d, 16, 32);
    if (lane < 16) s_dslf[rr] = d;
  }
  lds_fence();

  v8f acc[8];
  #pragma unroll
  for (int t = 0; t < 8; ++t) acc[t] = (v8f)0.f;
  float attsum_reg = 0.f;

  // ======== single-pass neighbor loop, double-buffered async gather ========
  for (int s = 0; s < S_N; ++s) {
    const float* neighS = smem + nbufOff[s & 1];

    if (s + 1 < S_N) {   // stage next slot while current computes
      for (int r = 0; r < TB; ++r) {
        int b = bBase + r; if (b >= B_TOT) b = B_TOT - 1;
        async_row_load(nbufOff[(s+1) & 1] + r*ROW_PAD,
                       neigh_table + (size_t)neigh_idx[b*S_N + s + 1] * F_DIM,
                       lane);
      }
      wait_async_16();   // current slot resident; next slot in flight
    } else {
      wait_async_0();
    }

    // n2 and neighbor-attention logit per row
    {
      const float* row = neighS + rr*ROW_PAD + half*64;
      const float* av  = s_aneigh + half*64;
      float d = 0.f, n2 = 0.f;
      for (int f = 0; f < 64; ++f) { float v = row[f]; d += v*av[f]; n2 += v*v; }
      d  += __shfl_xor(d, 16, 32);
      n2 += __shfl_xor(n2, 16, 32);
      if (lane < 16) {
        float a = __expf(fmaxf(s_dslf[rr] + d, 0.f));  // exp(relu(logit))
        attsum_reg += a;
        s_att[rr] = a;
        s_n2[rr]  = n2;
      }
    }
    lds_fence();

    // cross(16x16) = neigh(16x128) @ center^T(128x16)   [32 fp32 WMMAs]
    v8f cr = (v8f)0.f;
    for (int k0 = 0; k0 < F_DIM; k0 += 4) {
      v2f a = *(const v2f*)(neighS + rr*ROW_PAD + k0 + koff);
      v2f bf;
      bf.x = s_centerT[(k0+koff  )*16 + rr];
      bf.y = s_centerT[(k0+koff+1)*16 + rr];
      cr = __builtin_amdgcn_wmma_f32_16x16x4_f32(false, a, false, bf,
                                                 (short)0, cr, false, false);
    }

    // q' = att[m] / (n2[m] - 2*cross + c2[n] + 1); relayout C->A via LDS
    {
      float c2n = s_c2[rr];
      #pragma unroll
      for (int v = 0; v < 8; ++v) {
        int m = v + hi8;
        float q = s_att[m] / (s_n2[m] - 2.f*cr[v] + c2n + 1.f);
        qst[m*Q_PAD + rr] = q;
      }
    }
    lds_fence();

    // qcm(16x128) = q'(16x16) @ cluster_mask(16x128); acc += qcm * neigh
    #pragma unroll
    for (int t = 0; t < 8; ++t) {
      v8f c = (v8f)0.f;
      #pragma unroll
      for (int k0 = 0; k0 < K_CL; k0 += 4) {
        v2f a = *(const v2f*)(qst + rr*Q_PAD + k0 + koff);
        v2f bf;
        bf.x = s_cmask[(k0+koff  )*F_DIM + t*16 + rr];
        bf.y = s_cmask[(k0+koff+1)*F_DIM + t*16 + rr];
        c = __builtin_amdgcn_wmma_f32_16x16x4_f32(false, a, false, bf,
                                                  (short)0, c, false, false);
      }
      #pragma unroll
      for (int v = 0; v < 8; ++v) {
        float nv = neighS[(v + hi8)*ROW_PAD + t*16 + rr];
        acc[t][v] += c[v] * nv;
      }
    }
  }

  // ---------------- normalize by sum(att), stash agg into LDS ----------------
  if (lane < 16) s_asum[rr] = attsum_reg;
  lds_fence();
  #pragma unroll
  for (int t = 0; t < 8; ++t) {
    #pragma unroll
    for (int v = 0; v < 8; ++v) {
      int m = v + hi8;
      aggS[m*ROW_PAD + t*16 + rr] = acc[t][v] / s_asum[m];
    }
  }
  lds_fence();

  // -------------- out(16x128) = [self|agg](16x256) @ weight^T --------------
  for (int t = 0; t < 8; ++t) {
    v8f o = (v8f)0.f;
    for (int k0 = 0; k0 < 256; k0 += 4) {
      const float* arow = (k0 < 128) ? (selfF + rr*ROW_PAD + k0 + koff)
                                     : (aggS  + rr*ROW_PAD + k0 - 128 + koff);
      v2f a = *(const v2f*)arow;
      v2f bf;
      bf.x = s_wT[(k0+koff  )*128 + t*16 + rr];
      bf.y = s_wT[(k0+koff+1)*128 + t*16 + rr];
      o = __builtin_amdgcn_wmma_f32_16x16x4_f32(false, a, false, bf,
                                                (short)0, o, false, false);
    }
    #pragma unroll
    for (int v = 0; v < 8; ++v) {
      int b = bBase + v + hi8;
      if (b < B_TOT)
        out[(size_t)b*E_DIM + t*16 + rr] = fmaxf(o[v], 0.f);
    }
  }
}

extern "C" void kernel_launch(void* const* d_in, const int* in_sizes, int n_in,
                              void* d_out, int out_size, void* d_ws, size_t ws_size,
                              hipStream_t stream) {
  const int*   nodes        = (const int*)  d_in[0];
  const int*   neigh_idx    = (const int*)  d_in[1];
  const float* self_table   = (const float*)d_in[2];
  const float* neigh_table  = (const float*)d_in[3];
  const float* center       = (const float*)d_in[4];
  const float* cluster_mask = (const float*)d_in[5];
  const float* weight       = (const float*)d_in[6];
  const float* alpha        = (const float*)d_in[7];
  float* out = (float*)d_out;

  static_assert((size_t)SM_TOTAL * sizeof(float) <= 320u*1024u, "LDS budget");
  const size_t smem = (size_t)SM_TOTAL * sizeof(float);
  (void)hipFuncSetAttribute((const void*)clustered_attn_gnn_kernel,
                            hipFuncAttributeMaxDynamicSharedMemorySize, (int)smem);

  const int rowsPerWG = WAVES * TB;
  dim3 grid((B_TOT + rowsPerWG - 1) / rowsPerWG);
  clustered_attn_gnn_kernel<<<grid, WAVES*32, smem, stream>>>(
      nodes, neigh_idx, self_table, neigh_table, center, cluster_mask,
      weight, alpha, out);
  (void)in_sizes; (void)n_in; (void)out_size; (void)d_ws; (void)ws_size;
}